// Encoder_17669495455857
// MI455X (gfx1250) — compile-verified
//
#include <hip/hip_runtime.h>
#include <hip/hip_bf16.h>
#include <stdint.h>

// ---------------- problem constants (match reference) ----------------
#define NUM_GRAPHS      16
#define NODES_PER_GRAPH 4096
#define NTOT            (NUM_GRAPHS * NODES_PER_GRAPH)   // 65536
#define IN_FEAT         16
#define HIDDEN          32
#define FC1_OUT         256
#define FC2_OUT         64
#define NEDGE           1048576
#define KFC1            (NODES_PER_GRAPH * HIDDEN)       // 131072
#define FC1_KCHUNKS     256                              // split-K for FC1
#define FC1_TILE_K      (KFC1 / FC1_KCHUNKS)             // 512
#define FC1_LDS_STRIDE  516   // 516 % 64 == 4 -> lanes m=0..15 hit distinct banks

typedef __attribute__((ext_vector_type(2))) float v2f;
typedef __attribute__((ext_vector_type(8))) float v8f;
typedef __attribute__((ext_vector_type(4))) int   v4i;
typedef v4i __attribute__((address_space(1))) gv4i;   // global int32x4
typedef v4i __attribute__((address_space(3))) lv4i;   // LDS int32x4

// ---------------------------------------------------------------------
// Degree / normalization kernels
// ---------------------------------------------------------------------
__global__ void k_init_deg(float* __restrict__ deg) {
    int i = blockIdx.x * blockDim.x + threadIdx.x;
    if (i < NTOT) deg[i] = 1.0f;   // self-loop
}

__global__ void k_count_deg(const long long* __restrict__ ei, float* __restrict__ deg) {
    int e = blockIdx.x * blockDim.x + threadIdx.x;
    if (e < NEDGE) {
        int d = (int)ei[NEDGE + e];
        atomicAdd(&deg[d], 1.0f);
    }
}

__global__ void k_dinv(float* __restrict__ deg) {
    int i = blockIdx.x * blockDim.x + threadIdx.x;
    if (i < NTOT) deg[i] = rsqrtf(deg[i]);   // in-place: deg becomes dinv
}

// ---------------------------------------------------------------------
// Dense node GEMM: H[NTOT x HIDDEN] = X[NTOT x K] @ W[K x HIDDEN]
// One wave per 16x16 tile, f32 WMMA 16x16x4 (full precision).
// W (<=4KB) is staged once per block in LDS.
// A layout (f32 16x4): lanes 0-15 hold M=lane, VGPR j = K = 2*half + j.
// B/C/D: rows striped across lanes, lanes 16-31 carry the +2K / +8M split.
// ---------------------------------------------------------------------
template <int K>
__global__ void k_node_gemm(const float* __restrict__ X, const float* __restrict__ W,
                            float* __restrict__ H) {
    __shared__ float Ws[K * HIDDEN];
    for (int i = threadIdx.x; i < K * HIDDEN; i += blockDim.x) Ws[i] = W[i];
    __syncthreads();

    const int wave = (blockIdx.x * blockDim.x + threadIdx.x) >> 5;
    const int lane = threadIdx.x & 31;
    const int ntc  = HIDDEN / 16;            // 2 column tiles
    const int rowTile = wave / ntc;
    const int colTile = wave % ntc;
    if (rowTile >= NTOT / 16) return;
    const int row0 = rowTile * 16;
    const int col0 = colTile * 16;
    const int m    = lane & 15;
    const int half = lane >> 4;

    v8f c = {};
    for (int k0 = 0; k0 < K; k0 += 4) {
        const int kk = k0 + 2 * half;
        v2f a, b;
        a.x = X[(row0 + m) * K + kk];
        a.y = X[(row0 + m) * K + kk + 1];
        b.x = Ws[kk * HIDDEN + col0 + m];
        b.y = Ws[(kk + 1) * HIDDEN + col0 + m];
        c = __builtin_amdgcn_wmma_f32_16x16x4_f32(false, a, false, b, (short)0, c,
                                                  false, false);
    }
#pragma unroll
    for (int r = 0; r < 8; ++r)
        H[(row0 + r + 8 * half) * HIDDEN + col0 + m] = c[r];
}

// ---------------------------------------------------------------------
// agg initialized with the self-loop term: agg = h * dinv^2
// ---------------------------------------------------------------------
__global__ void k_self_init(const float* __restrict__ H, const float* __restrict__ dinv,
                            float* __restrict__ AGG) {
    int i = blockIdx.x * blockDim.x + threadIdx.x;
    if (i < NTOT * HIDDEN) {
        float s = dinv[i >> 5];  // node = i / HIDDEN
        AGG[i] = H[i] * s * s;
    }
}

// ---------------------------------------------------------------------
// Edge scatter: agg[dst] += h[src] * dinv[src]*dinv[dst]
// h rows are 128B -> eight float4 L2-resident gathers + f32 atomics.
// ---------------------------------------------------------------------
__global__ void k_scatter(const long long* __restrict__ ei, const float* __restrict__ H,
                          const float* __restrict__ dinv, float* __restrict__ AGG) {
    int e = blockIdx.x * blockDim.x + threadIdx.x;
    if (e >= NEDGE) return;
    int s = (int)ei[e];
    int d = (int)ei[NEDGE + e];
    float norm = dinv[s] * dinv[d];
    const float4* hs = (const float4*)(H + s * HIDDEN);
    float* ad = AGG + d * HIDDEN;
#pragma unroll
    for (int q = 0; q < HIDDEN / 4; ++q) {
        float4 v = hs[q];
        atomicAdd(&ad[4 * q + 0], v.x * norm);
        atomicAdd(&ad[4 * q + 1], v.y * norm);
        atomicAdd(&ad[4 * q + 2], v.z * norm);
        atomicAdd(&ad[4 * q + 3], v.w * norm);
    }
}

__global__ void k_bias_relu(const float* __restrict__ AGG, const float* __restrict__ b,
                            float* __restrict__ OUT) {
    int i = blockIdx.x * blockDim.x + threadIdx.x;
    if (i < NTOT * HIDDEN) {
        float v = AGG[i] + b[i & (HIDDEN - 1)];
        OUT[i] = v > 0.0f ? v : 0.0f;
    }
}

__global__ void k_zero(float* __restrict__ p, int n) {
    int i = blockIdx.x * blockDim.x + threadIdx.x;
    if (i < n) p[i] = 0.0f;
}

// ---------------------------------------------------------------------
// FC1: ACC[16 x 256] += H[16 x 131072] @ W[131072 x 256], split-K.
// One block (16 waves) per K-chunk of 512. The block stages its 16x512
// A-chunk into LDS ONCE (coalesced, async-to-LDS when available), then
// each wave owns one of the 16 column tiles: the 134MB weight matrix is
// streamed exactly once overall and A is read once instead of 16x.
// Partial tiles reduce via global f32 atomics.
// ---------------------------------------------------------------------
__global__ void __launch_bounds__(512) k_fc1(const float* __restrict__ H,
                                             const float* __restrict__ W,
                                             float* __restrict__ ACC) {
    __shared__ float As[NUM_GRAPHS * FC1_LDS_STRIDE];   // ~33KB

    const int kc   = blockIdx.x;          // 0..FC1_KCHUNKS-1
    const int kbeg = kc * FC1_TILE_K;
    const int tid  = threadIdx.x;

    // ---- cooperative stage: H[0..15][kbeg..kbeg+512) -> LDS (row stride 516)
    // 2048 float4 transfers, 4 per thread, fully coalesced on global side.
    for (int i = tid; i < NUM_GRAPHS * (FC1_TILE_K / 4); i += 512) {
        const int row = i / (FC1_TILE_K / 4);
        const int c4  = i % (FC1_TILE_K / 4);
        const float* gsrc = H + row * KFC1 + kbeg + c4 * 4;
        float*       ldst = &As[row * FC1_LDS_STRIDE + c4 * 4];
#if __has_builtin(__builtin_amdgcn_global_load_async_to_lds_b128)
        __builtin_amdgcn_global_load_async_to_lds_b128(
            (gv4i*)gsrc, (lv4i*)ldst, 0, 0);
#else
        *(float4*)ldst = *(const float4*)gsrc;
#endif
    }
#if __has_builtin(__builtin_amdgcn_s_wait_asynccnt)
    __builtin_amdgcn_s_wait_asynccnt(0);
#endif
    __syncthreads();

    // ---- WMMA phase: wave w -> column tile w
    const int wave = tid >> 5;            // 0..15
    const int lane = tid & 31;
    const int col0 = wave * 16;
    const int m    = lane & 15;
    const int half = lane >> 4;

    v8f c = {};
    for (int k0 = 0; k0 < FC1_TILE_K; k0 += 4) {
        const int kk = k0 + 2 * half;
        const int gk = kbeg + kk;
        // prefetch the weight stream ahead (emits global_prefetch_b8)
        __builtin_prefetch(&W[(gk + 64) * FC1_OUT + col0 + m], 0, 1);
        v2f a, b;
        a.x = As[m * FC1_LDS_STRIDE + kk];         // ds_load_b64, bank-clean
        a.y = As[m * FC1_LDS_STRIDE + kk + 1];
        b.x = W[gk * FC1_OUT + col0 + m];
        b.y = W[(gk + 1) * FC1_OUT + col0 + m];
        c = __builtin_amdgcn_wmma_f32_16x16x4_f32(false, a, false, b, (short)0, c,
                                                  false, false);
    }
#pragma unroll
    for (int r = 0; r < 8; ++r)
        atomicAdd(&ACC[(r + 8 * half) * FC1_OUT + col0 + m], c[r]);
}

__global__ void k_fc1_finish(const float* __restrict__ ACC, const float* __restrict__ b,
                             float* __restrict__ OUT) {
    int i = blockIdx.x * blockDim.x + threadIdx.x;
    if (i < NUM_GRAPHS * FC1_OUT) {
        float v = ACC[i] + b[i % FC1_OUT];
        OUT[i] = v > 0.0f ? v : 0.0f;
    }
}

// ---------------------------------------------------------------------
// FC2: OUT[16 x 64] = A[16 x 256] @ W[256 x 64] + b. One block, 4 waves.
// ---------------------------------------------------------------------
__global__ void k_fc2(const float* __restrict__ A, const float* __restrict__ W,
                      const float* __restrict__ bias, float* __restrict__ OUT) {
    const int wave = threadIdx.x >> 5;        // 0..3 -> column tile
    const int lane = threadIdx.x & 31;
    const int col0 = wave * 16;
    const int m    = lane & 15;
    const int half = lane >> 4;

    v8f c = {};
    for (int k0 = 0; k0 < FC1_OUT; k0 += 4) {
        const int kk = k0 + 2 * half;
        v2f a, b;
        a.x = A[m * FC1_OUT + kk];
        a.y = A[m * FC1_OUT + kk + 1];
        b.x = W[kk * FC2_OUT + col0 + m];
        b.y = W[(kk + 1) * FC2_OUT + col0 + m];
        c = __builtin_amdgcn_wmma_f32_16x16x4_f32(false, a, false, b, (short)0, c,
                                                  false, false);
    }
#pragma unroll
    for (int r = 0; r < 8; ++r)
        OUT[(r + 8 * half) * FC2_OUT + col0 + m] = c[r] + bias[col0 + m];
}

// ---------------------------------------------------------------------
// Launch
// ---------------------------------------------------------------------
extern "C" void kernel_launch(void* const* d_in, const int* in_sizes, int n_in,
                              void* d_out, int out_size, void* d_ws, size_t ws_size,
                              hipStream_t stream) {
    const float*     x     = (const float*)d_in[0];
    const long long* ei    = (const long long*)d_in[1];   // int64 [2, E]
    const float*     W1    = (const float*)d_in[2];
    const float*     b1    = (const float*)d_in[3];
    const float*     W2    = (const float*)d_in[4];
    const float*     b2    = (const float*)d_in[5];
    const float*     W3    = (const float*)d_in[6];
    const float*     b3    = (const float*)d_in[7];
    const float*     fc1_w = (const float*)d_in[8];
    const float*     fc1_b = (const float*)d_in[9];
    const float*     fc2_w = (const float*)d_in[10];
    const float*     fc2_b = (const float*)d_in[11];
    float*           out   = (float*)d_out;

    // workspace layout
    float* deg  = (float*)d_ws;                 // NTOT (becomes dinv)
    float* hA   = deg  + NTOT;                  // NTOT*HIDDEN
    float* hB   = hA   + NTOT * HIDDEN;         // NTOT*HIDDEN
    float* agg  = hB   + NTOT * HIDDEN;         // NTOT*HIDDEN
    float* acc1 = agg  + NTOT * HIDDEN;         // 16*256
    float* act1 = acc1 + NUM_GRAPHS * FC1_OUT;  // 16*256

    const int T = 256;
    const int gN   = (NTOT + T - 1) / T;
    const int gE   = (NEDGE + T - 1) / T;
    const int gNH  = (NTOT * HIDDEN + T - 1) / T;

    // normalization
    k_init_deg<<<gN, T, 0, stream>>>(deg);
    k_count_deg<<<gE, T, 0, stream>>>(ei, deg);
    k_dinv<<<gN, T, 0, stream>>>(deg);

    // GCN conv tiling: (NTOT/16 row tiles) * 2 col tiles waves, 8 waves/block
    const int gemmBlocks = ((NTOT / 16) * (HIDDEN / 16)) / (T / 32);

    // ---- conv1 ----
    k_node_gemm<IN_FEAT><<<gemmBlocks, T, 0, stream>>>(x, W1, hA);
    k_self_init<<<gNH, T, 0, stream>>>(hA, deg, agg);
    k_scatter<<<gE, T, 0, stream>>>(ei, hA, deg, agg);
    k_bias_relu<<<gNH, T, 0, stream>>>(agg, b1, hB);

    // ---- conv2 ----
    k_node_gemm<HIDDEN><<<gemmBlocks, T, 0, stream>>>(hB, W2, hA);
    k_self_init<<<gNH, T, 0, stream>>>(hA, deg, agg);
    k_scatter<<<gE, T, 0, stream>>>(ei, hA, deg, agg);
    k_bias_relu<<<gNH, T, 0, stream>>>(agg, b2, hB);

    // ---- conv3 ----
    k_node_gemm<HIDDEN><<<gemmBlocks, T, 0, stream>>>(hB, W3, hA);
    k_self_init<<<gNH, T, 0, stream>>>(hA, deg, agg);
    k_scatter<<<gE, T, 0, stream>>>(ei, hA, deg, agg);
    k_bias_relu<<<gNH, T, 0, stream>>>(agg, b3, hB);  // hB == pooled [16 x 131072]

    // ---- FC1 (LDS-staged A, split-K WMMA + atomic reduce) ----
    k_zero<<<(NUM_GRAPHS * FC1_OUT + T - 1) / T, T, 0, stream>>>(acc1, NUM_GRAPHS * FC1_OUT);
    k_fc1<<<FC1_KCHUNKS, 512, 0, stream>>>(hB, fc1_w, acc1);
    k_fc1_finish<<<(NUM_GRAPHS * FC1_OUT + T - 1) / T, T, 0, stream>>>(acc1, fc1_b, act1);

    // ---- FC2 ----
    k_fc2<<<1, 128, 0, stream>>>(act1, fc2_w, fc2_b, out);
}